// Graph_Transformer_multi_channel_88175678587413
// MI455X (gfx1250) — compile-verified
//
#include <hip/hip_runtime.h>
#include <hip/hip_bf16.h>

typedef __attribute__((ext_vector_type(2))) float v2f;
typedef __attribute__((ext_vector_type(8))) float v8f;

#define F_IN   20
#define F_OUT  8
#define EMBED  24
#define QKV3   72
#define HIN    84
#define H1     16
#define NCLASS 2

__device__ __forceinline__ v8f wmma_f32_16x16x4(v2f a, v2f b, v8f c) {
  // D = A(16x4 f32) * B(4x16 f32) + C(16x16 f32), full f32 precision
  return __builtin_amdgcn_wmma_f32_16x16x4_f32(false, a, false, b, (short)0, c,
                                               false, false);
}

// ---------------------------------------------------------------- zero
__global__ void k_zero(float* p, long n) {
  long i = (long)blockIdx.x * blockDim.x + threadIdx.x;
  long stride = (long)gridDim.x * blockDim.x;
  for (; i < n; i += stride) p[i] = 0.0f;
}

// ------------------------------------------------- dense transform x@w
// support[n, c*8+f] = sum_k x_c[n,k] * w_c[k,f]   (K=20 -> 5 WMMA k-steps)
__global__ void __launch_bounds__(256) k_transform(
    const float* __restrict__ x1, const float* __restrict__ x2,
    const float* __restrict__ x3, const float* __restrict__ w1,
    const float* __restrict__ w2, const float* __restrict__ w3,
    float* __restrict__ support, int ntiles) {
  const int lane = threadIdx.x & 31;
  const int wave = threadIdx.x >> 5;
  const int tile = blockIdx.x * 8 + wave;
  if (tile >= ntiles) return;            // wave-uniform: EXEC stays all-ones
  const int base = tile * 16;
  const int m0 = lane & 15;
  const int khalf = (lane < 16) ? 0 : 2;
  const bool cok = m0 < F_OUT;
  const int wcol = m0 & (F_OUT - 1);     // clamped column: load always valid
  const float* xs[3] = {x1, x2, x3};
  const float* ws_[3] = {w1, w2, w3};
#pragma unroll
  for (int c = 0; c < 3; ++c) {
    const float* x = xs[c];
    const float* w = ws_[c];
    v8f acc = {};
#pragma unroll
    for (int kb = 0; kb < F_IN; kb += 4) {
      int k0 = kb + khalf;
      v2f a, b;
      a.x = x[(long)(base + m0) * F_IN + k0];
      a.y = x[(long)(base + m0) * F_IN + k0 + 1];
      float t0 = w[k0 * F_OUT + wcol];        // unconditional load
      float t1 = w[(k0 + 1) * F_OUT + wcol];  // + cndmask select (no branch)
      b.x = cok ? t0 : 0.0f;
      b.y = cok ? t1 : 0.0f;
      acc = wmma_f32_16x16x4(a, b, acc);
    }
    if (cok) {
#pragma unroll
      for (int r = 0; r < 8; ++r) {
        int row = base + ((lane < 16) ? r : (8 + r));
        support[(long)row * EMBED + c * F_OUT + m0] = acc[r];
      }
    }
  }
}

// ------------------------------------------- edge gather/scale/scatter
// 8 lanes per edge, 3 channels fused: 3 f32 atomics per lane, edge data
// read once (77 MB HBM); support/acc stay L2-resident (19 MB each).
__global__ void __launch_bounds__(256) k_edges(
    const int* __restrict__ erow, const int* __restrict__ ecol,
    const float* __restrict__ eval_, const float* __restrict__ support,
    float* acc, long E) {
  long gid = (long)blockIdx.x * blockDim.x + threadIdx.x;
  long e = gid >> 3;
  if (e >= E) return;
  int f = (int)(gid & 7);
  int r = erow[e], c = ecol[e];
  float v = eval_[e];
  const float* s = support + (long)c * EMBED + f;
  float* a = acc + (long)r * EMBED + f;
  atomicAdd(a + 0,  s[0]  * v);
  atomicAdd(a + 8,  s[8]  * v);
  atomicAdd(a + 16, s[16] * v);
}

// -------------------------- relu(agg+b) -> x ; q/k/v WMMA ; qk partial
__global__ void __launch_bounds__(256) k_qkv(
    const float* acc,  // aliased with vbuf (block-local rows only)
    const float* __restrict__ b1, const float* __restrict__ b2,
    const float* __restrict__ b3, const float* __restrict__ wqkv,
    const float* __restrict__ bqkv, float* vbuf, float* qk, int ntiles) {
  __shared__ float xs[8][16][EMBED];
  __shared__ float qs[128 * EMBED];
  __shared__ float ks[128 * EMBED];
  const int lane = threadIdx.x & 31;
  const int wave = threadIdx.x >> 5;
  const int tile = blockIdx.x * 8 + wave;
  const bool valid = tile < ntiles;
  const int base = tile * 16;
  // phase A: stage x = relu(agg + bias) into LDS (invalid waves zero q/k)
  if (valid) {
    for (int j = lane; j < 16 * EMBED; j += 32) {
      int node = j / EMBED, d = j % EMBED;
      int ch = d >> 3, fi = d & 7;
      float bias = (ch == 0) ? b1[fi] : (ch == 1 ? b2[fi] : b3[fi]);
      float val = acc[(long)(base + node) * EMBED + d] + bias;
      xs[wave][node][d] = val > 0.0f ? val : 0.0f;
    }
  } else {
    for (int j = lane; j < 16 * EMBED; j += 32) {
      qs[wave * 16 * EMBED + j] = 0.0f;
      ks[wave * 16 * EMBED + j] = 0.0f;
    }
  }
  __syncthreads();
  // phase B: q,k,v = x @ wqkv + b  (K=24, N=24 as 16+8 tiles)
  if (valid) {
    const int m0 = lane & 15;
    const int khalf = (lane < 16) ? 0 : 2;
#pragma unroll
    for (int t = 0; t < 3; ++t) {  // 0=q, 1=k, 2=v
      int off = t * EMBED;
#pragma unroll
      for (int ct = 0; ct < 2; ++ct) {
        int ncol = ct * 16 + m0;
        bool cok = ncol < EMBED;
        int ncc = cok ? ncol : 0;  // clamped: load always in-bounds
        v8f cacc = {};
#pragma unroll
        for (int kb = 0; kb < EMBED; kb += 4) {
          int k0 = kb + khalf;
          v2f a, b;
          a.x = xs[wave][m0][k0];
          a.y = xs[wave][m0][k0 + 1];
          float t0 = wqkv[k0 * QKV3 + off + ncc];
          float t1 = wqkv[(k0 + 1) * QKV3 + off + ncc];
          b.x = cok ? t0 : 0.0f;
          b.y = cok ? t1 : 0.0f;
          cacc = wmma_f32_16x16x4(a, b, cacc);
        }
        if (cok) {
          float bias = bqkv[off + ncol];
#pragma unroll
          for (int r = 0; r < 8; ++r) {
            int M = (lane < 16) ? r : (8 + r);
            float val = cacc[r] + bias;
            if (t == 0)      qs[(wave * 16 + M) * EMBED + ncol] = val;
            else if (t == 1) ks[(wave * 16 + M) * EMBED + ncol] = val;
            else             vbuf[(long)(base + M) * EMBED + ncol] = val;
          }
        }
      }
    }
  }
  __syncthreads();
  // phase C: qk[d,e] += sum_n q[n,d]*k[n,e] over this block's 128 nodes
  for (int o = threadIdx.x; o < EMBED * EMBED; o += 256) {
    int d = o / EMBED, e2 = o % EMBED;
    float s = 0.0f;
    for (int n = 0; n < 128; ++n)
      s += qs[n * EMBED + d] * ks[n * EMBED + e2];
    atomicAdd(&qk[o], s);
  }
}

// -------------------------------------------------- softmax over [24,24]
__global__ void k_softmax(const float* __restrict__ qk, float* __restrict__ att) {
  int d = threadIdx.x;
  if (d >= EMBED) return;
  const float scale = 0.2041241452319315f;  // 1/sqrt(24)
  float row[EMBED];
  float m = -3.0e38f;
  for (int e = 0; e < EMBED; ++e) {
    row[e] = qk[d * EMBED + e] * scale;
    m = fmaxf(m, row[e]);
  }
  float s = 0.0f;
  for (int e = 0; e < EMBED; ++e) {
    row[e] = __expf(row[e] - m);
    s += row[e];
  }
  float inv = 1.0f / s;
  for (int e = 0; e < EMBED; ++e) att[d * EMBED + e] = row[e] * inv;
}

// ------------- out = v @ att^T ; h=[x1|x2|x3|out] ; MLP (WMMA + scalar)
__device__ __forceinline__ float h_elem(const float* x1, const float* x2,
                                        const float* x3,
                                        const float o[16][EMBED], int row,
                                        int node, int k) {
  if (k < 20) return x1[(long)row * F_IN + k];
  if (k < 40) return x2[(long)row * F_IN + (k - 20)];
  if (k < 60) return x3[(long)row * F_IN + (k - 40)];
  return o[node][k - 60];
}

__global__ void __launch_bounds__(256) k_final(
    const float* __restrict__ x1, const float* __restrict__ x2,
    const float* __restrict__ x3, const float* __restrict__ vbuf,
    const float* __restrict__ att, const float* __restrict__ wl1,
    const float* __restrict__ bl1, const float* __restrict__ wl2,
    const float* __restrict__ bl2, float* __restrict__ outp, int ntiles) {
  __shared__ float att_s[EMBED * EMBED];
  __shared__ float wl1_s[HIN * H1];
  __shared__ float outs[8][16][EMBED];
  __shared__ float h1s[8][16][H1];
  for (int i = threadIdx.x; i < EMBED * EMBED; i += 256) att_s[i] = att[i];
  for (int i = threadIdx.x; i < HIN * H1; i += 256) wl1_s[i] = wl1[i];
  __syncthreads();
  const int lane = threadIdx.x & 31;
  const int wave = threadIdx.x >> 5;
  const int tile = blockIdx.x * 8 + wave;
  const bool valid = tile < ntiles;
  const int base = tile * 16;
  const int m0 = lane & 15;
  const int khalf = (lane < 16) ? 0 : 2;
  // out[n,d] = sum_e v[n,e] * att[d,e]  -> B[k,n] = att_s[n*24 + k]
  if (valid) {
#pragma unroll
    for (int ct = 0; ct < 2; ++ct) {
      int ncol = ct * 16 + m0;
      bool cok = ncol < EMBED;
      int ncc = cok ? ncol : 0;  // clamped LDS address
      v8f cacc = {};
#pragma unroll
      for (int kb = 0; kb < EMBED; kb += 4) {
        int k0 = kb + khalf;
        v2f a, b;
        a.x = vbuf[(long)(base + m0) * EMBED + k0];
        a.y = vbuf[(long)(base + m0) * EMBED + k0 + 1];
        float t0 = att_s[ncc * EMBED + k0];
        float t1 = att_s[ncc * EMBED + k0 + 1];
        b.x = cok ? t0 : 0.0f;
        b.y = cok ? t1 : 0.0f;
        cacc = wmma_f32_16x16x4(a, b, cacc);
      }
      if (cok) {
#pragma unroll
        for (int r = 0; r < 8; ++r) {
          int M = (lane < 16) ? r : (8 + r);
          outs[wave][M][ncol] = cacc[r];
        }
      }
    }
  }
  __syncthreads();
  // layer 1: relu([x1|x2|x3|out] @ wl1 + bl1)  (K=84 -> 21 WMMA)
  if (valid) {
    v8f c1 = {};
#pragma unroll
    for (int kb = 0; kb < HIN; kb += 4) {
      int k0 = kb + khalf;
      v2f a, b;
      a.x = h_elem(x1, x2, x3, outs[wave], base + m0, m0, k0);
      a.y = h_elem(x1, x2, x3, outs[wave], base + m0, m0, k0 + 1);
      b.x = wl1_s[k0 * H1 + m0];
      b.y = wl1_s[(k0 + 1) * H1 + m0];
      c1 = wmma_f32_16x16x4(a, b, c1);
    }
    float bias = bl1[m0];
#pragma unroll
    for (int r = 0; r < 8; ++r) {
      int M = (lane < 16) ? r : (8 + r);
      float val = c1[r] + bias;
      h1s[wave][M][m0] = val > 0.0f ? val : 0.0f;
    }
  }
  __syncthreads();
  // layer 2: relu(h1 @ wl2 + bl2), N=2 -> scalar (one output per lane)
  if (valid) {
    int node = lane >> 1;
    int cls = lane & 1;
    float s = bl2[cls];
#pragma unroll
    for (int j = 0; j < H1; ++j) s += h1s[wave][node][j] * wl2[j * NCLASS + cls];
    s = s > 0.0f ? s : 0.0f;
    outp[(long)(base + node) * NCLASS + cls] = s;
  }
}

extern "C" void kernel_launch(void* const* d_in, const int* in_sizes, int n_in,
                              void* d_out, int out_size, void* d_ws,
                              size_t ws_size, hipStream_t stream) {
  const float* x1 = (const float*)d_in[0];
  const float* x2 = (const float*)d_in[1];
  const float* x3 = (const float*)d_in[2];
  const int* erow = (const int*)d_in[3];
  const int* ecol = (const int*)d_in[4];
  const float* ev = (const float*)d_in[5];
  const float* w1 = (const float*)d_in[6];
  const float* b1 = (const float*)d_in[7];
  const float* w2 = (const float*)d_in[8];
  const float* b2 = (const float*)d_in[9];
  const float* w3 = (const float*)d_in[10];
  const float* b3 = (const float*)d_in[11];
  const float* wqkv = (const float*)d_in[12];
  const float* bqkv = (const float*)d_in[13];
  const float* wl1 = (const float*)d_in[14];
  const float* bl1 = (const float*)d_in[15];
  const float* wl2 = (const float*)d_in[16];
  const float* bl2 = (const float*)d_in[17];

  const int N = in_sizes[0] / F_IN;   // 200000 (multiple of 16)
  const long E = in_sizes[3];         // 6400000
  const int ntiles = (N + 15) / 16;   // 12500
  const int blocks = (ntiles + 7) / 8;

  float* ws = (float*)d_ws;
  float* support = ws;                       // [N,24]
  float* accv = ws + (size_t)N * EMBED;      // [N,24]; acc, then reused for v
  float* qk = accv + (size_t)N * EMBED;      // [576]
  float* att = qk + EMBED * EMBED;           // [576]

  // acc + qk are contiguous: zero both in one pass
  k_zero<<<2048, 256, 0, stream>>>(accv, (long)N * EMBED + EMBED * EMBED);
  k_transform<<<blocks, 256, 0, stream>>>(x1, x2, x3, w1, w2, w3, support,
                                          ntiles);
  long ethreads = E * 8;
  int eblocks = (int)((ethreads + 255) / 256);
  k_edges<<<eblocks, 256, 0, stream>>>(erow, ecol, ev, support, accv, E);
  k_qkv<<<blocks, 256, 0, stream>>>(accv, b1, b2, b3, wqkv, bqkv, accv, qk,
                                    ntiles);
  k_softmax<<<1, 32, 0, stream>>>(qk, att);
  k_final<<<blocks, 256, 0, stream>>>(x1, x2, x3, accv, att, wl1, bl1, wl2,
                                      bl2, (float*)d_out, ntiles);
}